// GraphConv_12120397709959
// MI455X (gfx1250) — compile-verified
//
#include <hip/hip_runtime.h>
#include <hip/hip_bf16.h>

#define N_NODES 100000
#define N_EDGES 1600000
#define IN_DIM  64
#define OUT_DIM 64

typedef float v2f __attribute__((ext_vector_type(2)));
typedef float v8f __attribute__((ext_vector_type(8)));

// ---------------------------------------------------------------------------
// Kernel 1: zero the aggregation workspace (float4 stores, fully coalesced)
// ---------------------------------------------------------------------------
__global__ void gc_zero_kernel(float4* __restrict__ p, int n4) {
    int i = blockIdx.x * blockDim.x + threadIdx.x;
    if (i < n4) {
        float4 z = {0.0f, 0.0f, 0.0f, 0.0f};
        p[i] = z;
    }
}

// ---------------------------------------------------------------------------
// Kernel 2: edge scatter  agg[dst] += val * x[src]
// One wave32 per edge; each lane owns 2 consecutive features.
// x and agg are both L2-resident (25.6 MB each vs 192 MB L2), so the
// hardware f32 atomics (global_atomic_add_f32) resolve in L2.
// ---------------------------------------------------------------------------
__global__ void gc_scatter_kernel(const float* __restrict__ x,
                                  const int*   __restrict__ esrc,
                                  const int*   __restrict__ edst,
                                  const float* __restrict__ eval,
                                  float*       __restrict__ agg) {
    const int gtid = blockIdx.x * blockDim.x + threadIdx.x;
    const int edge = gtid >> 5;
    const int lane = gtid & 31;
    if (edge >= N_EDGES) return;

    const int   src = esrc[edge];
    const int   dst = edst[edge];
    const float val = eval[edge];

    const v2f* xr = (const v2f*)(x + (size_t)src * IN_DIM) + lane;
    v2f v = *xr;

    float* ar = agg + (size_t)dst * IN_DIM + (lane << 1);
    unsafeAtomicAdd(ar,     val * v.x);   // global_atomic_add_f32
    unsafeAtomicAdd(ar + 1, val * v.y);   // global_atomic_add_f32
}

// ---------------------------------------------------------------------------
// Kernel 3: out = agg @ W^T + b  via V_WMMA_F32_16X16X4_F32 (full f32)
// One wave per 16-node tile. K=64 -> 16 chained WMMAs per 16-wide N tile,
// 4 N tiles cover OUT_DIM=64.
//
// A 16x4 f32 fragment (ISA 7.12.2): lane L holds
//   A[M = L%16, K = k0 + 2*(L/16) + {0,1}]        (one v2f, 8B load)
// B 4x16 fragment (B = W^T, so B[k][n] = W[n][k]): lane L holds
//   W[nt*16 + L%16, k0 + 2*(L/16) + {0,1}]        (one v2f, 8B load)
// D 16x16 f32: VGPR j, lane L -> M = j + 8*(L/16), N = L%16.
// ---------------------------------------------------------------------------
__global__ void gc_gemm_wmma_kernel(const float* __restrict__ agg,
                                    const float* __restrict__ W,
                                    const float* __restrict__ bias,
                                    float*       __restrict__ out) {
    const int lane  = threadIdx.x & 31;
    const int wave  = threadIdx.x >> 5;
    const int tile  = blockIdx.x * (blockDim.x >> 5) + wave;
    const int nTiles = N_NODES / 16;          // 6250, exact
    if (tile >= nTiles) return;               // wave-uniform

    const int half = lane >> 4;               // 0: lanes 0-15, 1: lanes 16-31
    const int l16  = lane & 15;
    const int row_base = tile * 16;

    // Hoist the A fragments for the whole K=64 row segment (32 VGPRs/lane).
    const float* arow = agg + (size_t)(row_base + l16) * IN_DIM + 2 * half;
    v2f afrag[16];
#pragma unroll
    for (int k = 0; k < 16; ++k)
        afrag[k] = *(const v2f*)(arow + 4 * k);

#pragma unroll
    for (int nt = 0; nt < 4; ++nt) {
        const float* wrow = W + (size_t)(nt * 16 + l16) * IN_DIM + 2 * half;
        v8f c = {};
#pragma unroll
        for (int k = 0; k < 16; ++k) {
            v2f bb = *(const v2f*)(wrow + 4 * k);
            c = __builtin_amdgcn_wmma_f32_16x16x4_f32(
                    /*neg_a=*/false, afrag[k],
                    /*neg_b=*/false, bb,
                    /*c_mod=*/(short)0, c,
                    /*reuse_a=*/false, /*reuse_b=*/false);
        }
        const float bv = bias[nt * 16 + l16];
        float* obase = out + (size_t)(row_base + 8 * half) * OUT_DIM + nt * 16 + l16;
#pragma unroll
        for (int j = 0; j < 8; ++j)
            obase[(size_t)j * OUT_DIM] = c[j] + bv;
    }
}

// ---------------------------------------------------------------------------
extern "C" void kernel_launch(void* const* d_in, const int* in_sizes, int n_in,
                              void* d_out, int out_size, void* d_ws, size_t ws_size,
                              hipStream_t stream) {
    const float* x    = (const float*)d_in[0];
    const int*   esrc = (const int*)  d_in[1];
    const int*   edst = (const int*)  d_in[2];
    const float* eval = (const float*)d_in[3];
    const float* W    = (const float*)d_in[4];
    const float* b    = (const float*)d_in[5];
    float*       out  = (float*)d_out;
    float*       agg  = (float*)d_ws;         // needs N_NODES*IN_DIM*4 = 25.6 MB

    // 1) agg = 0
    {
        const int n4 = N_NODES * IN_DIM / 4;  // 1.6M float4
        gc_zero_kernel<<<(n4 + 255) / 256, 256, 0, stream>>>((float4*)agg, n4);
    }
    // 2) scatter-add: one wave per edge
    {
        const long long threads = (long long)N_EDGES * 32;
        const int blocks = (int)((threads + 255) / 256);
        gc_scatter_kernel<<<blocks, 256, 0, stream>>>(x, esrc, edst, eval, agg);
    }
    // 3) WMMA GEMM + bias: one wave per 16-row tile, 8 waves per block
    {
        const int nTiles = N_NODES / 16;      // 6250
        const int wavesPerBlock = 8;
        const int blocks = (nTiles + wavesPerBlock - 1) / wavesPerBlock;
        gc_gemm_wmma_kernel<<<blocks, wavesPerBlock * 32, 0, stream>>>(agg, W, b, out);
    }
}